// SA_module_32298154066795
// MI455X (gfx1250) — compile-verified
//
#include <hip/hip_runtime.h>
#include <hip/hip_bf16.h>
#include <float.h>

#define B_    4
#define N_    8192
#define C_    64
#define OUT_  64
#define KNN_  16

typedef __attribute__((ext_vector_type(2))) float v2f;
typedef __attribute__((ext_vector_type(8))) float v8f;

// ------------------------------------------------------------------
// Kernel 1: 16-NN per query. One thread per query; xyz_s staged in LDS
// as float4 (single ds_load_b128 per candidate); register-resident
// sorted top-16; prefetch of next tile overlaps the scan.
// ------------------------------------------------------------------
__global__ __launch_bounds__(256) void knn_kernel(const float* __restrict__ xyz,
                                                  const float* __restrict__ xyz_s,
                                                  int* __restrict__ idx_out) {
  const int b = blockIdx.y;
  const int q = blockIdx.x * 256 + threadIdx.x;
  const float* xb  = xyz   + (size_t)b * 3 * N_;
  const float* xsb = xyz_s + (size_t)b * 3 * N_;

  const float qx = xb[0 * N_ + q];
  const float qy = xb[1 * N_ + q];
  const float qz = xb[2 * N_ + q];

  float bd[KNN_];
  int   bi[KNN_];
#pragma unroll
  for (int j = 0; j < KNN_; ++j) { bd[j] = FLT_MAX; bi[j] = 0; }

  __shared__ float4 st[256];

  for (int base = 0; base < N_; base += 256) {
    const float tx = xsb[0 * N_ + base + threadIdx.x];
    const float ty = xsb[1 * N_ + base + threadIdx.x];
    const float tz = xsb[2 * N_ + base + threadIdx.x];
    st[threadIdx.x] = make_float4(tx, ty, tz, 0.0f);
    __syncthreads();

    if (base + 256 < N_) {  // overlap next tile's fetch with this scan
      __builtin_prefetch(&xsb[0 * N_ + base + 256 + threadIdx.x], 0, 1);
      __builtin_prefetch(&xsb[1 * N_ + base + 256 + threadIdx.x], 0, 1);
      __builtin_prefetch(&xsb[2 * N_ + base + 256 + threadIdx.x], 0, 1);
    }

#pragma unroll 4
    for (int t = 0; t < 256; ++t) {
      const float4 s = st[t];
      const float dx = qx - s.x;
      const float dy = qy - s.y;
      const float dz = qz - s.z;
      const float d2 = fmaf(dx, dx, fmaf(dy, dy, dz * dz));
      if (d2 < bd[KNN_ - 1]) {            // common case: one compare, skip
        float dc = d2; int ic = base + t; // fully unrolled insertion
#pragma unroll
        for (int j = 0; j < KNN_; ++j) {
          const bool sw = dc < bd[j];
          const float td = bd[j]; const int ti = bi[j];
          if (sw) { bd[j] = dc; bi[j] = ic; dc = td; ic = ti; }
        }
      }
    }
    __syncthreads();
  }
  int* op = idx_out + ((size_t)b * N_ + q) * KNN_;
#pragma unroll
  for (int j = 0; j < KNN_; ++j) op[j] = bi[j];
}

// ------------------------------------------------------------------
// Kernel 2: outT[b][n][o] = sum_c W[o][wcol0+c] * X[b][c][n] (+ bias[o])
// via V_WMMA_F32_16X16X4_F32 (f32 full precision). One wave per
// 16(row)x16(col) tile, K=64 as 16 chained WMMAs. Output point-major
// (N,64) so the downstream neighbor gather is contiguous per point.
// ------------------------------------------------------------------
__global__ __launch_bounds__(128) void gemm_wmma_kernel(const float* __restrict__ W, int ldw, int wcol0,
                                                        const float* __restrict__ X,
                                                        const float* __restrict__ bias,
                                                        float* __restrict__ outT) {
  const int b     = blockIdx.y;
  const int nBase = blockIdx.x * 16;
  const int wave  = threadIdx.x >> 5;   // 0..3 -> row tile
  const int lane  = threadIdx.x & 31;
  const int mBase = wave * 16;
  const int nl    = lane & 15;
  const int half  = lane >> 4;

  const float* Xb   = X + (size_t)b * C_ * N_;
  const float* wrow = W + (size_t)(mBase + nl) * ldw + wcol0;

  v8f acc = {};
#pragma unroll
  for (int k0 = 0; k0 < C_; k0 += 4) {
    const int ka = k0 + 2 * half;
    v2f av, bv;
    av.x = wrow[ka];
    av.y = wrow[ka + 1];
    bv.x = Xb[(size_t)ka * N_ + nBase + nl];
    bv.y = Xb[(size_t)(ka + 1) * N_ + nBase + nl];
    acc = __builtin_amdgcn_wmma_f32_16x16x4_f32(
        /*neg_a=*/false, av, /*neg_b=*/false, bv,
        /*c_mod=*/(short)0, acc, /*reuse_a=*/false, /*reuse_b=*/false);
  }

  const int oBase = mBase + half * 8;
  float* po = outT + ((size_t)b * N_ + nBase + nl) * OUT_ + oBase;
#pragma unroll
  for (int r = 0; r < 8; ++r) {
    float v = acc[r];
    if (bias) v += bias[oBase + r];
    po[r] = v;
  }
}

// ------------------------------------------------------------------
// Kernel 3: fused geo features + g-branch 1x1 conv + modulation + max_k.
// One wave per query point; lane handles out channels {lane, lane+32}.
// f = relu(P1T[n] + P2T[m]);  g = relu(Wg . geo10 + bg);  out = max_k g*f
// Epilogue staged through LDS for coalesced (channel-major) stores.
// ------------------------------------------------------------------
__global__ __launch_bounds__(256) void fuse_kernel(const float* __restrict__ xyz,
                                                   const float* __restrict__ xyz_s,
                                                   const int* __restrict__ idx,
                                                   const float* __restrict__ Wg,
                                                   const float* __restrict__ bg,
                                                   const float* __restrict__ P1T,
                                                   const float* __restrict__ P2T,
                                                   float* __restrict__ out) {
  const int b     = blockIdx.y;
  const int wave  = threadIdx.x >> 5;
  const int lane  = threadIdx.x & 31;
  const int nBase = blockIdx.x * 8;
  const int n     = nBase + wave;

  const float* xb  = xyz   + (size_t)b * 3 * N_;
  const float* xsb = xyz_s + (size_t)b * 3 * N_;
  const float qx = xb[0 * N_ + n];
  const float qy = xb[1 * N_ + n];
  const float qz = xb[2 * N_ + n];

  const int o0 = lane, o1 = lane + 32;
  float wg0[10], wg1[10];
#pragma unroll
  for (int c = 0; c < 10; ++c) {
    wg0[c] = Wg[o0 * 10 + c];
    wg1[c] = Wg[o1 * 10 + c];
  }
  const float bg0 = bg[o0], bg1 = bg[o1];

  const float* p1 = P1T + ((size_t)b * N_ + n) * OUT_;
  const float f10 = p1[o0], f11 = p1[o1];

  // preload all 16 neighbor indices as 4x global_load_b128
  const int4* ip4 = (const int4*)(idx + ((size_t)b * N_ + n) * KNN_);
  int mm[KNN_];
#pragma unroll
  for (int j = 0; j < 4; ++j) {
    const int4 iv = ip4[j];
    mm[4 * j + 0] = iv.x; mm[4 * j + 1] = iv.y;
    mm[4 * j + 2] = iv.z; mm[4 * j + 3] = iv.w;
  }

  float acc0 = -FLT_MAX, acc1 = -FLT_MAX;
#pragma unroll
  for (int k = 0; k < KNN_; ++k) {
    const int m = mm[k];
    const float sxv = xsb[0 * N_ + m];
    const float syv = xsb[1 * N_ + m];
    const float szv = xsb[2 * N_ + m];
    const float rx = qx - sxv, ry = qy - syv, rz = qz - szv;
    const float d = sqrtf(fmaf(rx, rx, fmaf(ry, ry, rz * rz)));

    float g0 = bg0, g1 = bg1;
    g0 = fmaf(wg0[0], d,   g0); g1 = fmaf(wg1[0], d,   g1);
    g0 = fmaf(wg0[1], qx,  g0); g1 = fmaf(wg1[1], qx,  g1);
    g0 = fmaf(wg0[2], qy,  g0); g1 = fmaf(wg1[2], qy,  g1);
    g0 = fmaf(wg0[3], qz,  g0); g1 = fmaf(wg1[3], qz,  g1);
    g0 = fmaf(wg0[4], sxv, g0); g1 = fmaf(wg1[4], sxv, g1);
    g0 = fmaf(wg0[5], syv, g0); g1 = fmaf(wg1[5], syv, g1);
    g0 = fmaf(wg0[6], szv, g0); g1 = fmaf(wg1[6], szv, g1);
    g0 = fmaf(wg0[7], rx,  g0); g1 = fmaf(wg1[7], rx,  g1);
    g0 = fmaf(wg0[8], ry,  g0); g1 = fmaf(wg1[8], ry,  g1);
    g0 = fmaf(wg0[9], rz,  g0); g1 = fmaf(wg1[9], rz,  g1);
    g0 = fmaxf(g0, 0.0f);       g1 = fmaxf(g1, 0.0f);

    const float* p2 = P2T + ((size_t)b * N_ + m) * OUT_;
    const float f0 = fmaxf(f10 + p2[o0], 0.0f);
    const float f1 = fmaxf(f11 + p2[o1], 0.0f);

    acc0 = fmaxf(acc0, g0 * f0);
    acc1 = fmaxf(acc1, g1 * f1);
  }

  // stage through LDS so global stores are channel-major coalesced
  __shared__ float sOut[8][OUT_ + 1];   // +1 pitch: avoid 64-bank conflicts
  sOut[wave][o0] = acc0;
  sOut[wave][o1] = acc1;
  __syncthreads();

#pragma unroll
  for (int i = 0; i < 2; ++i) {
    const int v  = threadIdx.x + i * 256;  // 512 values: o*8 + nn
    const int o  = v >> 3;
    const int nn = v & 7;
    out[((size_t)b * OUT_ + o) * N_ + nBase + nn] = sOut[nn][o];
  }
}

// ------------------------------------------------------------------
extern "C" void kernel_launch(void* const* d_in, const int* in_sizes, int n_in,
                              void* d_out, int out_size, void* d_ws, size_t ws_size,
                              hipStream_t stream) {
  (void)in_sizes; (void)n_in; (void)out_size; (void)ws_size;
  const float* xyz   = (const float*)d_in[0];
  const float* xyz_s = (const float*)d_in[1];
  const float* fea   = (const float*)d_in[2];
  const float* fea_s = (const float*)d_in[3];
  const float* Wg    = (const float*)d_in[4];
  const float* bg    = (const float*)d_in[5];
  const float* Wf    = (const float*)d_in[6];
  const float* bf    = (const float*)d_in[7];
  float* out = (float*)d_out;

  char* ws = (char*)d_ws;
  int*   idx = (int*)ws;                                  // B*N*K ints   = 2 MB
  float* P1T = (float*)(ws + (size_t)B_ * N_ * KNN_ * 4); // B*N*64 f32   = 8 MB
  float* P2T = P1T + (size_t)B_ * N_ * OUT_;              // B*N*64 f32   = 8 MB

  // 1) KNN
  knn_kernel<<<dim3(N_ / 256, B_), 256, 0, stream>>>(xyz, xyz_s, idx);

  // 2) P1 = Wf[:, :64] * fea + bf ;  P2 = Wf[:, 64:] * fea_s   (f32 WMMA)
  gemm_wmma_kernel<<<dim3(N_ / 16, B_), 128, 0, stream>>>(Wf, 2 * C_, 0,  fea,   bf,      P1T);
  gemm_wmma_kernel<<<dim3(N_ / 16, B_), 128, 0, stream>>>(Wf, 2 * C_, C_, fea_s, nullptr, P2T);

  // 3) fused geo conv + modulation + max over neighbors
  fuse_kernel<<<dim3(N_ / 8, B_), 256, 0, stream>>>(xyz, xyz_s, idx, Wg, bg, P1T, P2T, out);
}